// GATModel_4947802325327
// MI455X (gfx1250) — compile-verified
//
#include <hip/hip_runtime.h>
#include <math.h>

typedef __attribute__((ext_vector_type(2))) float v2f;
typedef __attribute__((ext_vector_type(8))) float v8f;

#define GAT_H 4
#define NEG_SLOPE 0.2f

// ---------------------------------------------------------------------------
// generic fill
// ---------------------------------------------------------------------------
__global__ void fill_kernel(float* __restrict__ p, float v, int n) {
  int i = blockIdx.x * blockDim.x + threadIdx.x;
  if (i < n) p[i] = v;
}

// ---------------------------------------------------------------------------
// C[M x 256] = A[M x K] * B[K x 256]  via V_WMMA_F32_16X16X4_F32 (wave32)
// One block = 256 threads = 8 waves. Block owns a 16-row strip of A (LDS),
// each wave computes two 16x16 output tiles (16 N-tiles across 256 cols).
// A-fragment layout (ISA 7.12.2, 32-bit A 16x4): lane m=lane&15, g=lane>>4,
// VGPR j holds K = k0 + 2g + j. C layout: VGPR v -> row v + 8g, col = lane&15.
// ---------------------------------------------------------------------------
__global__ void __launch_bounds__(256)
wmma_gemm_kernel(const float* __restrict__ A, const float* __restrict__ B,
                 float* __restrict__ C, int M, int K) {
  const int NCOL = 256;
  const int ld = K + 4;                    // pad: stride 132/68 -> conflict-free LDS
  __shared__ float As[16 * (128 + 4)];

  int m0 = blockIdx.x * 16;
  // cooperative, coalesced load of the 16xK A strip
  for (int i = threadIdx.x; i < 16 * K; i += 256) {
    int r = i / K, c = i - r * K;
    int gr = m0 + r;
    As[r * ld + c] = (gr < M) ? A[(size_t)gr * K + c] : 0.0f;
  }
  __syncthreads();

  int wave = threadIdx.x >> 5;
  int lane = threadIdx.x & 31;
  int mloc = lane & 15;
  int g    = lane >> 4;

  for (int t = 0; t < 2; ++t) {
    int n0 = (wave * 2 + t) * 16;
    v8f acc = {};
#pragma unroll 4
    for (int k0 = 0; k0 < K; k0 += 4) {
      int ka = k0 + 2 * g;
      v2f a, b;
      a.x = As[mloc * ld + ka];
      a.y = As[mloc * ld + ka + 1];
      b.x = B[(size_t)ka * NCOL + n0 + mloc];
      b.y = B[(size_t)(ka + 1) * NCOL + n0 + mloc];
      acc = __builtin_amdgcn_wmma_f32_16x16x4_f32(false, a, false, b,
                                                  (short)0, acc, false, false);
    }
    int col = n0 + mloc;
#pragma unroll
    for (int v = 0; v < 8; ++v) {
      int row = m0 + v + 8 * g;
      if (row < M) C[(size_t)row * NCOL + col] = acc[v];
    }
  }
}

// ---------------------------------------------------------------------------
// alpha_src[n,h] = <hfeat[n,h,:], a_s[h,:]> ; same for alpha_dst (C = 64)
// ---------------------------------------------------------------------------
__global__ void alpha_kernel(const float* __restrict__ hfeat,
                             const float* __restrict__ a_s,
                             const float* __restrict__ a_d,
                             float* __restrict__ alpha_s,
                             float* __restrict__ alpha_d, int N) {
  int idx = blockIdx.x * blockDim.x + threadIdx.x;
  if (idx >= N * GAT_H) return;
  int node = idx >> 2;
  int h    = idx & 3;
  const float* hf = hfeat + (size_t)node * 256 + h * 64;
  const float* as = a_s + h * 64;
  const float* ad = a_d + h * 64;
  float ss = 0.f, sd = 0.f;
#pragma unroll 8
  for (int c = 0; c < 64; ++c) {
    float v = hf[c];
    ss += v * as[c];
    sd += v * ad[c];
  }
  alpha_s[idx] = ss;
  alpha_d[idx] = sd;
}

// ---------------------------------------------------------------------------
// float atomic max via ordered-int trick (mbuf initialized to -inf)
// ---------------------------------------------------------------------------
__device__ __forceinline__ void atomicMaxF(float* addr, float val) {
  if (val >= 0.0f)
    atomicMax((int*)addr, __float_as_int(val));
  else
    atomicMin((unsigned int*)addr, __float_as_uint(val));
}

__device__ __forceinline__ float lrelu(float v) {
  return v >= 0.0f ? v : NEG_SLOPE * v;
}

// pass 1: edge logits + leaky relu + segment max over dst
__global__ void edge_logits_max(const int* __restrict__ src,
                                const int* __restrict__ dst,
                                const float* __restrict__ as,
                                const float* __restrict__ ad,
                                float* __restrict__ ebuf,
                                float* __restrict__ mbuf, int E) {
  int e = blockIdx.x * blockDim.x + threadIdx.x;
  if (e >= E) return;
  int s = src[e], d = dst[e];
  float4 a1 = *(const float4*)(as + (size_t)s * 4);
  float4 a2 = *(const float4*)(ad + (size_t)d * 4);
  float4 v;
  v.x = lrelu(a1.x + a2.x);
  v.y = lrelu(a1.y + a2.y);
  v.z = lrelu(a1.z + a2.z);
  v.w = lrelu(a1.w + a2.w);
  *(float4*)(ebuf + (size_t)e * 4) = v;
  float* mb = mbuf + (size_t)d * 4;
  atomicMaxF(mb + 0, v.x);
  atomicMaxF(mb + 1, v.y);
  atomicMaxF(mb + 2, v.z);
  atomicMaxF(mb + 3, v.w);
}

// pass 2: exp(e - max) + segment sum over dst
__global__ void edge_exp_sum(const int* __restrict__ dst,
                             float* __restrict__ ebuf,
                             const float* __restrict__ mbuf,
                             float* __restrict__ denom, int E) {
  int e = blockIdx.x * blockDim.x + threadIdx.x;
  if (e >= E) return;
  int d = dst[e];
  float4 v = *(const float4*)(ebuf + (size_t)e * 4);
  float4 m = *(const float4*)(mbuf + (size_t)d * 4);
  v.x = __expf(v.x - m.x);
  v.y = __expf(v.y - m.y);
  v.z = __expf(v.z - m.z);
  v.w = __expf(v.w - m.w);
  *(float4*)(ebuf + (size_t)e * 4) = v;
  float* dn = denom + (size_t)d * 4;
  atomicAdd(dn + 0, v.x);
  atomicAdd(dn + 1, v.y);
  atomicAdd(dn + 2, v.z);
  atomicAdd(dn + 3, v.w);
}

// pass 3: acc[dst, c] += mean_h( alpha[e,h] * hfeat[src, h, c] )
// 64 threads per edge (channel-parallel), head-mean fused -> 4x fewer atomics
__global__ void __launch_bounds__(256)
edge_aggregate(const int* __restrict__ src, const int* __restrict__ dst,
               const float* __restrict__ ebuf, const float* __restrict__ denom,
               const float* __restrict__ hfeat, float* __restrict__ acc, int E) {
  int t = blockIdx.x * 256 + threadIdx.x;
  int e = t >> 6;
  int c = t & 63;
  if (e >= E) return;
  int s = src[e], d = dst[e];
  float4 w  = *(const float4*)(ebuf + (size_t)e * 4);
  float4 dn = *(const float4*)(denom + (size_t)d * 4);
  float a0 = w.x / (dn.x + 1e-16f);
  float a1 = w.y / (dn.y + 1e-16f);
  float a2 = w.z / (dn.z + 1e-16f);
  float a3 = w.w / (dn.w + 1e-16f);
  const float* hf = hfeat + (size_t)s * 256;
  float sum = a0 * hf[c] + a1 * hf[64 + c] + a2 * hf[128 + c] + a3 * hf[192 + c];
  atomicAdd(acc + (size_t)d * 64 + c, 0.25f * sum);
}

// h = relu(h + b), b broadcast over 64 channels
__global__ void bias_relu(float* __restrict__ h, const float* __restrict__ b,
                          int total) {
  int i = blockIdx.x * blockDim.x + threadIdx.x;
  if (i >= total) return;
  float v = h[i] + b[i & 63];
  h[i] = v > 0.0f ? v : 0.0f;
}

// log_softmax over 64 channels; one wave32 per node, 2 channels/lane
__global__ void __launch_bounds__(256)
logsoftmax_kernel(const float* __restrict__ acc, const float* __restrict__ b,
                  float* __restrict__ out, int N) {
  int node = blockIdx.x * 8 + (threadIdx.x >> 5);
  int lane = threadIdx.x & 31;
  if (node >= N) return;
  float v0 = acc[(size_t)node * 64 + lane] + b[lane];
  float v1 = acc[(size_t)node * 64 + 32 + lane] + b[32 + lane];
  float m = fmaxf(v0, v1);
#pragma unroll
  for (int off = 16; off > 0; off >>= 1) m = fmaxf(m, __shfl_xor(m, off, 32));
  float s = expf(v0 - m) + expf(v1 - m);
#pragma unroll
  for (int off = 16; off > 0; off >>= 1) s += __shfl_xor(s, off, 32);
  float lse = logf(s);
  out[(size_t)node * 64 + lane]      = v0 - m - lse;
  out[(size_t)node * 64 + 32 + lane] = v1 - m - lse;
}

// ---------------------------------------------------------------------------
extern "C" void kernel_launch(void* const* d_in, const int* in_sizes, int n_in,
                              void* d_out, int out_size, void* d_ws, size_t ws_size,
                              hipStream_t stream) {
  const float* x   = (const float*)d_in[0];
  const int*   ei  = (const int*)d_in[1];
  const float* W1  = (const float*)d_in[2];
  const float* as1 = (const float*)d_in[3];
  const float* ad1 = (const float*)d_in[4];
  const float* b1  = (const float*)d_in[5];
  const float* W2  = (const float*)d_in[6];
  const float* as2 = (const float*)d_in[7];
  const float* ad2 = (const float*)d_in[8];
  const float* b2  = (const float*)d_in[9];

  const int N = in_sizes[0] / 128;
  const int E = in_sizes[1] / 2;
  const int* src = ei;
  const int* dst = ei + E;

  // workspace carve-up (all float32)
  float* ws    = (float*)d_ws;
  float* hfeat = ws;  ws += (size_t)N * 256;   // N x H x 64 features (both layers)
  float* h1    = ws;  ws += (size_t)N * 64;    // layer accum / relu'd features
  float* alps  = ws;  ws += (size_t)N * 4;
  float* alpd  = ws;  ws += (size_t)N * 4;
  float* mbuf  = ws;  ws += (size_t)N * 4;
  float* denom = ws;  ws += (size_t)N * 4;
  float* ebuf  = ws;  ws += (size_t)E * 4;
  (void)ws_size; (void)n_in; (void)out_size;

  const int TB = 256;
  dim3 blk(TB);
  int gM   = (N + 15) / 16;                 // GEMM strips
  int gNH  = (N * 4 + TB - 1) / TB;
  int gN4  = gNH;
  int gN64 = (N * 64 + TB - 1) / TB;
  int gE   = (E + TB - 1) / TB;
  int gE64 = ((size_t)E * 64 + TB - 1) / TB;
  int gLS  = (N + 7) / 8;

  for (int layer = 0; layer < 2; ++layer) {
    const float* A   = (layer == 0) ? x   : h1;
    const float* W   = (layer == 0) ? W1  : W2;
    const float* a_s = (layer == 0) ? as1 : as2;
    const float* a_d = (layer == 0) ? ad1 : ad2;
    const int    K   = (layer == 0) ? 128 : 64;

    // dense projection: hfeat = A @ W   (WMMA fp32)
    wmma_gemm_kernel<<<gM, blk, 0, stream>>>(A, W, hfeat, N, K);
    // per-node attention halves
    alpha_kernel<<<gNH, blk, 0, stream>>>(hfeat, a_s, a_d, alps, alpd, N);
    // reset segment-softmax state + accumulator (h1 free after GEMM read it)
    fill_kernel<<<gN4, blk, 0, stream>>>(mbuf, -INFINITY, N * 4);
    fill_kernel<<<gN4, blk, 0, stream>>>(denom, 0.0f, N * 4);
    fill_kernel<<<gN64, blk, 0, stream>>>(h1, 0.0f, N * 64);
    // edge phases
    edge_logits_max<<<gE, blk, 0, stream>>>(src, dst, alps, alpd, ebuf, mbuf, E);
    edge_exp_sum<<<gE, blk, 0, stream>>>(dst, ebuf, mbuf, denom, E);
    edge_aggregate<<<gE64, blk, 0, stream>>>(src, dst, ebuf, denom, hfeat, h1, E);

    if (layer == 0) {
      bias_relu<<<gN64, blk, 0, stream>>>(h1, b1, N * 64);
    } else {
      logsoftmax_kernel<<<gLS, blk, 0, stream>>>(h1, b2, (float*)d_out, N);
    }
  }
}